// PathwayGraphEmbedding_11184094839169
// MI455X (gfx1250) — compile-verified
//
#include <hip/hip_runtime.h>

#define BN 32
#define NN 2000
#define EE 64000
#define DD 128   // DIN == DH == 128

typedef __attribute__((ext_vector_type(2))) float v2f;
typedef __attribute__((ext_vector_type(8))) float v8f;

// ---------------------------------------------------------------- degree counts
__global__ void k_deg(const int* __restrict__ dst, int* __restrict__ counts) {
  int e = blockIdx.x * blockDim.x + threadIdx.x;
  if (e < EE) atomicAdd(&counts[dst[e]], 1);
}

// single-wave exclusive scan of counts[0..NN) -> offsets[0..NN]
__global__ void k_scan(const int* __restrict__ counts, int* __restrict__ offsets) {
  const int lane = threadIdx.x & 31;
  const int CH = 63;                      // 32*63 = 2016 >= NN
  int base = lane * CH;
  int s = 0;
  for (int j = 0; j < CH; j++) {
    int idx = base + j;
    s += (idx < NN) ? counts[idx] : 0;
  }
  int inc = s;
  for (int d = 1; d < 32; d <<= 1) {
    int v = __shfl_up(inc, d, 32);
    if (lane >= d) inc += v;
  }
  int run = inc - s;                      // exclusive prefix of this lane's chunk
  for (int j = 0; j < CH; j++) {
    int idx = base + j;
    if (idx < NN) { offsets[idx] = run; run += counts[idx]; }
  }
  if (lane == 31) offsets[NN] = run;      // total edge count
}

__global__ void k_selfnorm(const int* __restrict__ counts, float* __restrict__ dinv,
                           float* __restrict__ selfn) {
  int i = blockIdx.x * blockDim.x + threadIdx.x;
  if (i < NN) {
    float d = (float)counts[i] + 1.0f;
    float r = rsqrtf(d);
    dinv[i]  = r;
    selfn[i] = r * r;
  }
}

__global__ void k_norme(const int* __restrict__ src, const int* __restrict__ dst,
                        const float* __restrict__ dinv, float* __restrict__ norme) {
  int e = blockIdx.x * blockDim.x + threadIdx.x;
  if (e < EE) norme[e] = dinv[src[e]] * dinv[dst[e]];
}

// CSR column build: bin edge ids by destination
__global__ void k_scatter(const int* __restrict__ dst, const int* __restrict__ offsets,
                          int* __restrict__ cursor, int* __restrict__ csr) {
  int e = blockIdx.x * blockDim.x + threadIdx.x;
  if (e < EE) {
    int d = dst[e];
    int pos = offsets[d] + atomicAdd(&cursor[d], 1);
    csr[pos] = e;
  }
}

// ---------------------------------------------------------------- fp32 WMMA GEMM
// H[64000x128] = X[64000x128] @ W[128x128], per-wave 16x128 strip, 8 waves/block.
__global__ __launch_bounds__(256) void k_gemm_wmma(const float* __restrict__ X,
                                                   const float* __restrict__ W,
                                                   float* __restrict__ H) {
  __shared__ float sW[DD * DD];  // 64 KB, XOR-swizzled on column bit5 by row bit1
  const int tid = threadIdx.x;

  // Async-stage W into LDS (CDNA5 ASYNCcnt path): 16 B per lane per iteration.
  for (int i = tid; i < (DD * DD) / 4; i += 256) {
    int r  = i >> 5;            // row (i*4 / 128)
    int c4 = (i & 31) << 2;     // column of this float4
    int cs = c4 ^ ((r & 2) << 4);   // swizzle keeps half-wave B reads bank-disjoint
    unsigned lds_off = (unsigned)(uintptr_t)(&sW[r * DD + cs]);
    const float* gp = W + (i << 2);
    asm volatile("global_load_async_to_lds_b128 %0, %1, off"
                 :: "v"(lds_off), "v"(gp) : "memory");
  }
  asm volatile("s_wait_asynccnt 0x0" ::: "memory");
  __syncthreads();

  const int wave  = tid >> 5;
  const int lane  = tid & 31;
  const int l16   = lane & 15;
  const int khalf = (lane >> 4) << 1;          // 0 for lanes 0-15, 2 for lanes 16-31
  const long rowBase = (long)blockIdx.x * 128 + wave * 16;

  v8f acc[8] = {};                              // 8 column tiles of 16x16 f32

  const float* Xrow = X + (rowBase + l16) * DD;
  const int swz = khalf << 4;
  for (int k = 0; k < DD; k += 4) {
    // A fragment (16x4 f32): lane m holds K = khalf, khalf+1
    float2 av = *reinterpret_cast<const float2*>(Xrow + k + khalf);
    v2f a = {av.x, av.y};
#pragma unroll
    for (int n = 0; n < 8; n++) {
      // B fragment (4x16 f32): VGPR0 = row k+khalf, VGPR1 = row k+khalf+1
      int ccol = (n * 16 + l16) ^ swz;
      v2f b = {sW[(k + khalf) * DD + ccol], sW[(k + khalf + 1) * DD + ccol]};
      acc[n] = __builtin_amdgcn_wmma_f32_16x16x4_f32(
          false, a, false, b, (short)0, acc[n], false, false);
    }
  }

  // C/D layout: VGPR v -> rows (v, v+8) split across half-waves, N = lane%16
  const int mAdd = (lane >> 4) << 3;
#pragma unroll
  for (int n = 0; n < 8; n++) {
#pragma unroll
    for (int v = 0; v < 8; v++) {
      long row = rowBase + mAdd + v;
      H[row * DD + n * 16 + l16] = acc[n][v];
    }
  }
}

// ---------------------------------------------------------------- fused GCN aggregation
// One wave per (batch, node): y[b,i,:] = relu?( sum_{e: dst=i} h[b,src_e,:]*norm_e
//                                               + h[b,i,:]*selfn[i] + bias )
__global__ __launch_bounds__(256) void k_gather(const float* __restrict__ h,
                                                const int* __restrict__ csr,
                                                const int* __restrict__ offsets,
                                                const int* __restrict__ esrc,
                                                const float* __restrict__ norme,
                                                const float* __restrict__ selfn,
                                                const float* __restrict__ bias,
                                                float* __restrict__ y, int do_relu) {
  int w    = (blockIdx.x * blockDim.x + threadIdx.x) >> 5;  // node id over B*N (exact fit)
  int lane = threadIdx.x & 31;                              // float4 chunk: dims lane*4..+3
  int b = w / NN;
  int i = w - b * NN;
  const float4* h4 = reinterpret_cast<const float4*>(h);

  float4 bv = reinterpret_cast<const float4*>(bias)[lane];
  float4 hv = h4[(long)w * 32 + lane];
  float  sn = selfn[i];
  float4 acc = {hv.x * sn + bv.x, hv.y * sn + bv.y, hv.z * sn + bv.z, hv.w * sn + bv.w};

  int jbeg = offsets[i], jend = offsets[i + 1];
  long brow = (long)b * NN;
  for (int j = jbeg; j < jend; j++) {
    int e  = csr[j];
    float ne = norme[e];
    int s  = esrc[e];
    float4 m = h4[(brow + s) * 32 + lane];
    acc.x += m.x * ne; acc.y += m.y * ne; acc.z += m.z * ne; acc.w += m.w * ne;
  }
  if (do_relu) {
    acc.x = fmaxf(acc.x, 0.f); acc.y = fmaxf(acc.y, 0.f);
    acc.z = fmaxf(acc.z, 0.f); acc.w = fmaxf(acc.w, 0.f);
  }
  reinterpret_cast<float4*>(y)[(long)w * 32 + lane] = acc;
}

// ---------------------------------------------------------------- mean pool
__global__ void k_mean(const float* __restrict__ y, float* __restrict__ out) {
  int b = blockIdx.x;      // 32 blocks
  int d = threadIdx.x;     // 128 threads
  float s = 0.f;
  for (int i = 0; i < NN; i++) s += y[((long)b * NN + i) * DD + d];
  out[b * DD + d] = s * (1.0f / NN);
}

// ---------------------------------------------------------------- launch
extern "C" void kernel_launch(void* const* d_in, const int* in_sizes, int n_in,
                              void* d_out, int out_size, void* d_ws, size_t ws_size,
                              hipStream_t stream) {
  const float* gene = (const float*)d_in[0];
  const int*   esrc = (const int*)d_in[1];
  const int*   edst = (const int*)d_in[2];
  const float* W1   = (const float*)d_in[3];
  const float* b1   = (const float*)d_in[4];
  const float* W2   = (const float*)d_in[5];
  const float* b2   = (const float*)d_in[6];
  float* out = (float*)d_out;
  float* ws  = (float*)d_ws;

  // workspace layout
  float* selfn = ws;                        // 2048 floats
  float* dinv  = ws + 2048;                 // 2048
  float* norme = ws + 4096;                 // 64512 (rounded)
  float* h     = ws + 4096 + 64512;         // 8,192,000
  float* y     = h + (long)BN * NN * DD;    // 8,192,000
  int* counts  = (int*)(y + (long)BN * NN * DD);  // 2048 ints
  int* cursor  = counts + 2048;             // 2048
  int* offsets = cursor + 2048;             // 2048 (need NN+1)
  int* csr     = offsets + 2048;            // 64000

  const int R = BN * NN;                    // 64000 rows
  const int gthreads = (BN * NN * 32) / 256; // gather blocks: 8000

  hipMemsetAsync(counts, 0, 4096 * sizeof(int), stream);  // counts + cursor
  k_deg<<<(EE + 255) / 256, 256, 0, stream>>>(edst, counts);
  k_scan<<<1, 32, 0, stream>>>(counts, offsets);
  k_selfnorm<<<(NN + 255) / 256, 256, 0, stream>>>(counts, dinv, selfn);
  k_norme<<<(EE + 255) / 256, 256, 0, stream>>>(esrc, edst, dinv, norme);
  k_scatter<<<(EE + 255) / 256, 256, 0, stream>>>(edst, offsets, cursor, csr);

  // ---- layer 1
  k_gemm_wmma<<<R / 128, 256, 0, stream>>>(gene, W1, h);
  k_gather<<<gthreads, 256, 0, stream>>>(h, csr, offsets, esrc, norme, selfn, b1, y, 1);

  // ---- layer 2 (h buffer reused for h2; y overwritten after GEMM consumes it)
  k_gemm_wmma<<<R / 128, 256, 0, stream>>>(y, W2, h);
  k_gather<<<gthreads, 256, 0, stream>>>(h, csr, offsets, esrc, norme, selfn, b2, y, 0);

  k_mean<<<BN, DD, 0, stream>>>(y, out);
}